// GCNEncoder_47906065219800
// MI455X (gfx1250) — compile-verified
//
#include <hip/hip_runtime.h>
#include <hip/hip_bf16.h>

typedef float v2f __attribute__((ext_vector_type(2)));
typedef float v8f __attribute__((ext_vector_type(8)));

// ---------------------------------------------------------------------------
// Generic utility: zero-fill a float buffer
// ---------------------------------------------------------------------------
__global__ __launch_bounds__(256) void fill_zero_kernel(float* __restrict__ p, size_t n) {
    size_t i = (size_t)blockIdx.x * blockDim.x + threadIdx.x;
    if (i < n) p[i] = 0.0f;
}

// ---------------------------------------------------------------------------
// Degree count: deg[dst] += 1 for every edge (self-loops folded in later)
// ---------------------------------------------------------------------------
__global__ __launch_bounds__(256) void count_deg_kernel(const int* __restrict__ dst,
                                                        float* __restrict__ deg, int E) {
    int e = blockIdx.x * blockDim.x + threadIdx.x;
    if (e < E) atomicAdd(&deg[dst[e]], 1.0f);
}

// dinv[i] = rsqrt(deg[i] + 1)   (self-loop adds 1 -> deg >= 1 always)
__global__ __launch_bounds__(256) void dinv_kernel(float* __restrict__ deg, int n) {
    int i = blockIdx.x * blockDim.x + threadIdx.x;
    if (i < n) deg[i] = rsqrtf(deg[i] + 1.0f);
}

// ---------------------------------------------------------------------------
// FP32 WMMA GEMM:  C[M,N] = A[M,K] @ B[K,N] (+bias, optional ReLU)
// One wave -> one 16x16 tile via V_WMMA_F32_16X16X4_F32 (full fp32 precision).
// A layout per lane: lanes 0-15 hold K=k..k+1, lanes 16-31 hold K=k+2..k+3.
// Requires M%16==0, N%16==0, K%4==0 (true for all GEMMs in this model).
// ---------------------------------------------------------------------------
__global__ __launch_bounds__(256) void gemm_f32_wmma_kernel(
    const float* __restrict__ A, const float* __restrict__ B,
    const float* __restrict__ bias, float* __restrict__ C,
    int M, int K, int N, int relu)
{
    const int lane = threadIdx.x & 31;
    const int wave = threadIdx.x >> 5;
    const int row0 = blockIdx.x * 16;
    const int col0 = (blockIdx.y * 8 + wave) * 16;
    if (row0 >= M || col0 >= N) return;   // wave-uniform: EXEC stays all-1s

    const int  l15   = lane & 15;
    const int  khalf = (lane >> 4) * 2;   // 0 for lanes 0-15, 2 for lanes 16-31
    const float* arow = A + (size_t)(row0 + l15) * K;   // A row for this lane
    const float* bcol = B + col0 + l15;                 // B column for this lane

    v8f acc = {};
    #pragma unroll 4
    for (int k = 0; k < K; k += 4) {
        const int kk = k + khalf;
        v2f a = *(const v2f*)(arow + kk);               // global_load_b64
        v2f b;
        b.x = bcol[(size_t)kk * N];
        b.y = bcol[(size_t)(kk + 1) * N];
        acc = __builtin_amdgcn_wmma_f32_16x16x4_f32(
            /*neg_a=*/false, a, /*neg_b=*/false, b,
            /*c_mod=*/(short)0, acc, /*reuse_a=*/false, /*reuse_b=*/false);
    }

    const int   n  = col0 + l15;
    const float bv = bias ? bias[n] : 0.0f;
    const int   mb = row0 + ((lane >> 4) * 8);          // D layout: M split at lane 16
    #pragma unroll
    for (int r = 0; r < 8; ++r) {
        float v = acc[r] + bv;
        if (relu) v = fmaxf(v, 0.0f);
        C[(size_t)(mb + r) * N + n] = v;
    }
}

// ---------------------------------------------------------------------------
// Edge scatter: agg[dst] += dinv[src]*dinv[dst] * h[src]   (128 feats, F=128)
// One wave per edge: lane handles a float4 (32*4 = 128 features).
// src/dst are wave-uniform -> scalarized; gather is a coalesced 512B read.
// ---------------------------------------------------------------------------
__global__ __launch_bounds__(256) void edge_scatter_kernel(
    const int* __restrict__ src, const int* __restrict__ dst,
    const float* __restrict__ dinv, const float* __restrict__ h,
    float* __restrict__ agg, int E)
{
    const int e = blockIdx.x * 8 + (threadIdx.x >> 5);
    if (e >= E) return;
    const int lane = threadIdx.x & 31;
    const int s = src[e];
    const int d = dst[e];
    const float w = dinv[s] * dinv[d];
    const float4 hv = *(const float4*)(h + (size_t)s * 128 + lane * 4);
    float* out = agg + (size_t)d * 128 + lane * 4;
    atomicAdd(out + 0, w * hv.x);
    atomicAdd(out + 1, w * hv.y);
    atomicAdd(out + 2, w * hv.z);
    atomicAdd(out + 3, w * hv.w);
}

// ---------------------------------------------------------------------------
// Node finalize: act[i,f] = [relu]( agg[i,f] + dinv[i]^2 * h[i,f] + bias[f] )
// (adds the self-loop contribution and the layer bias; writes in place)
// ---------------------------------------------------------------------------
__global__ __launch_bounds__(256) void node_finalize_kernel(
    float* __restrict__ agg, const float* __restrict__ h,
    const float* __restrict__ dinv, const float* __restrict__ bias,
    int n_nodes, int relu)
{
    size_t idx = (size_t)blockIdx.x * blockDim.x + threadIdx.x;
    if (idx >= (size_t)n_nodes * 128) return;
    const int i = (int)(idx >> 7);        // /128
    const int f = (int)(idx & 127);
    const float di = dinv[i];
    float v = agg[idx] + di * di * h[idx] + bias[f];
    if (relu) v = fmaxf(v, 0.0f);
    agg[idx] = v;
}

// ---------------------------------------------------------------------------
// Pooling
// ---------------------------------------------------------------------------
__global__ __launch_bounds__(256) void count_batch_kernel(const int* __restrict__ batch,
                                                          float* __restrict__ counts, int n) {
    int i = blockIdx.x * blockDim.x + threadIdx.x;
    if (i < n) atomicAdd(&counts[batch[i]], 1.0f);
}

__global__ __launch_bounds__(256) void pool_scatter_kernel(
    const int* __restrict__ batch, const float* __restrict__ act,
    float* __restrict__ gsum, int n_nodes)
{
    const int i = blockIdx.x * 8 + (threadIdx.x >> 5);
    if (i >= n_nodes) return;
    const int lane = threadIdx.x & 31;
    const int b = batch[i];
    const float4 v = *(const float4*)(act + (size_t)i * 128 + lane * 4);
    float* out = gsum + (size_t)b * 128 + lane * 4;
    atomicAdd(out + 0, v.x);
    atomicAdd(out + 1, v.y);
    atomicAdd(out + 2, v.z);
    atomicAdd(out + 3, v.w);
}

__global__ __launch_bounds__(256) void pool_div_kernel(float* __restrict__ gsum,
                                                       const float* __restrict__ counts,
                                                       int n_graphs) {
    int idx = blockIdx.x * blockDim.x + threadIdx.x;
    if (idx < n_graphs * 128) gsum[idx] /= fmaxf(counts[idx >> 7], 1.0f);
}

// ---------------------------------------------------------------------------
// Host-side launcher
// ---------------------------------------------------------------------------
static inline int cdiv(long long a, long long b) { return (int)((a + b - 1) / b); }

extern "C" void kernel_launch(void* const* d_in, const int* in_sizes, int n_in,
                              void* d_out, int out_size, void* d_ws, size_t ws_size,
                              hipStream_t stream) {
    // Inputs (setup_inputs dict order)
    const float* x    = (const float*)d_in[0];
    const int*   ei   = (const int*)d_in[1];    // [2, E]
    const int*   bat  = (const int*)d_in[2];    // [N]
    const float* W0   = (const float*)d_in[3];
    const float* b0   = (const float*)d_in[4];
    const float* W1   = (const float*)d_in[5];
    const float* b1   = (const float*)d_in[6];
    const float* W2   = (const float*)d_in[7];
    const float* b2   = (const float*)d_in[8];
    const float* Wm1  = (const float*)d_in[9];
    const float* bm1  = (const float*)d_in[10];
    const float* Wm2  = (const float*)d_in[11];
    const float* bm2  = (const float*)d_in[12];

    const int NN   = in_sizes[2];            // 50000 nodes
    const int E    = in_sizes[1] / 2;        // 800000 edges
    const int FIN  = in_sizes[0] / NN;       // 64
    const int FH   = in_sizes[3] / FIN;      // 128
    const int NHID = in_sizes[9] / FH;       // 512
    const int NOUT = in_sizes[11] / NHID;    // 256
    const int NG   = out_size / NOUT;        // 256

    const int* src = ei;
    const int* dst = ei + E;

    // Workspace layout (floats), 256B-aligned slices
    float* ws   = (float*)d_ws;
    size_t off  = 0;
    float* dinv = ws + off; off += ((size_t)NN + 63) & ~63ull;          // deg -> dinv
    float* h    = ws + off; off += (size_t)NN * FH;                     // GEMM output
    float* agg  = ws + off; off += (size_t)NN * FH;                     // aggregation / act
    float* cnts = ws + off; off += 64;                                  // graph counts
    float* gsum = ws + off; off += (size_t)NG * FH;                     // pooled features
    float* g1   = ws + off; off += (size_t)NG * NHID;                   // MLP hidden
    (void)ws_size;

    const size_t nodeF = (size_t)NN * FH;

    // ---- GCN normalization -------------------------------------------------
    fill_zero_kernel<<<cdiv(NN, 256), 256, 0, stream>>>(dinv, NN);
    count_deg_kernel<<<cdiv(E, 256), 256, 0, stream>>>(dst, dinv, E);
    dinv_kernel<<<cdiv(NN, 256), 256, 0, stream>>>(dinv, NN);

    // ---- Layer 0: h = x @ W0 ; aggregate ; +b0 ; ReLU ----------------------
    {
        dim3 g(cdiv(NN, 16), cdiv(FH, 128));
        gemm_f32_wmma_kernel<<<g, 256, 0, stream>>>(x, W0, nullptr, h, NN, FIN, FH, 0);
    }
    fill_zero_kernel<<<cdiv((long long)nodeF, 256), 256, 0, stream>>>(agg, nodeF);
    edge_scatter_kernel<<<cdiv(E, 8), 256, 0, stream>>>(src, dst, dinv, h, agg, E);
    node_finalize_kernel<<<cdiv((long long)nodeF, 256), 256, 0, stream>>>(agg, h, dinv, b0, NN, 1);

    // ---- Layer 1 -----------------------------------------------------------
    {
        dim3 g(cdiv(NN, 16), cdiv(FH, 128));
        gemm_f32_wmma_kernel<<<g, 256, 0, stream>>>(agg, W1, nullptr, h, NN, FH, FH, 0);
    }
    fill_zero_kernel<<<cdiv((long long)nodeF, 256), 256, 0, stream>>>(agg, nodeF);
    edge_scatter_kernel<<<cdiv(E, 8), 256, 0, stream>>>(src, dst, dinv, h, agg, E);
    node_finalize_kernel<<<cdiv((long long)nodeF, 256), 256, 0, stream>>>(agg, h, dinv, b1, NN, 1);

    // ---- Layer 2 (no ReLU) -------------------------------------------------
    {
        dim3 g(cdiv(NN, 16), cdiv(FH, 128));
        gemm_f32_wmma_kernel<<<g, 256, 0, stream>>>(agg, W2, nullptr, h, NN, FH, FH, 0);
    }
    fill_zero_kernel<<<cdiv((long long)nodeF, 256), 256, 0, stream>>>(agg, nodeF);
    edge_scatter_kernel<<<cdiv(E, 8), 256, 0, stream>>>(src, dst, dinv, h, agg, E);
    node_finalize_kernel<<<cdiv((long long)nodeF, 256), 256, 0, stream>>>(agg, h, dinv, b2, NN, 0);

    // ---- Global mean pool --------------------------------------------------
    fill_zero_kernel<<<1, 256, 0, stream>>>(cnts, NG);
    fill_zero_kernel<<<cdiv((long long)NG * FH, 256), 256, 0, stream>>>(gsum, (size_t)NG * FH);
    count_batch_kernel<<<cdiv(NN, 256), 256, 0, stream>>>(bat, cnts, NN);
    pool_scatter_kernel<<<cdiv(NN, 8), 256, 0, stream>>>(bat, agg, gsum, NN);
    pool_div_kernel<<<cdiv((long long)NG * FH, 256), 256, 0, stream>>>(gsum, cnts, NG);

    // ---- MLP: Linear -> ReLU -> Linear ------------------------------------
    {
        dim3 g(cdiv(NG, 16), cdiv(NHID, 128));
        gemm_f32_wmma_kernel<<<g, 256, 0, stream>>>(gsum, Wm1, bm1, g1, NG, FH, NHID, 1);
    }
    {
        dim3 g(cdiv(NG, 16), cdiv(NOUT, 128));
        gemm_f32_wmma_kernel<<<g, 256, 0, stream>>>(g1, Wm2, bm2, (float*)d_out, NG, NHID, NOUT, 0);
    }
}